// ColorHistogramLoss_12704513261608
// MI455X (gfx1250) — compile-verified
//
#include <hip/hip_runtime.h>

#define NBINS 4096  // 16^3

typedef float v2f __attribute__((ext_vector_type(2)));
typedef float v4f __attribute__((ext_vector_type(4)));
typedef float v8f __attribute__((ext_vector_type(8)));

__device__ __forceinline__ int quant16(float c) {
    int q = (int)(c * 15.0f);          // truncation == astype(int32) for c >= 0
    q = q < 0 ? 0 : q;
    q = q > 15 ? 15 : q;
    return q;
}

__global__ void zero_hist_kernel(unsigned int* __restrict__ p, int n) {
    int i = blockIdx.x * blockDim.x + threadIdx.x;
    if (i < n) p[i] = 0u;
}

// Streaming source histogram: LDS-private 4096-bin histogram per block,
// 4 points (48B) per lane-iteration via three NT b128 loads.
__global__ __launch_bounds__(1024)
void hist_src_kernel(const float* __restrict__ colors,
                     unsigned int* __restrict__ ghist, int npoints) {
    __shared__ unsigned int lhist[NBINS];
    for (int i = threadIdx.x; i < NBINS; i += blockDim.x) lhist[i] = 0u;
    __syncthreads();

    const int ngroups = npoints >> 2;                   // 4 points per group
    const int gstride = gridDim.x * blockDim.x;
    for (int g = blockIdx.x * blockDim.x + threadIdx.x; g < ngroups; g += gstride) {
        const v4f* p = (const v4f*)(colors + (size_t)g * 12);
        // prefetch next stride's tile into the cache hierarchy (speculative)
        __builtin_prefetch((const void*)(colors + ((size_t)g + gstride) * 12), 0, 1);
        v4f a = __builtin_nontemporal_load(p + 0);
        v4f b = __builtin_nontemporal_load(p + 1);
        v4f c = __builtin_nontemporal_load(p + 2);
        int b0 = (quant16(a.x) * 16 + quant16(a.y)) * 16 + quant16(a.z);
        int b1 = (quant16(a.w) * 16 + quant16(b.x)) * 16 + quant16(b.y);
        int b2 = (quant16(b.z) * 16 + quant16(b.w)) * 16 + quant16(c.x);
        int b3 = (quant16(c.y) * 16 + quant16(c.z)) * 16 + quant16(c.w);
        atomicAdd(&lhist[b0], 1u);   // ds_add_u32
        atomicAdd(&lhist[b1], 1u);
        atomicAdd(&lhist[b2], 1u);
        atomicAdd(&lhist[b3], 1u);
    }
    // tail (npoints not divisible by 4) — first few global threads handle it
    {
        int base = ngroups << 2;
        int t = blockIdx.x * blockDim.x + threadIdx.x;
        if (base + t < npoints) {
            const float* p = colors + (size_t)(base + t) * 3;
            int bb = (quant16(p[0]) * 16 + quant16(p[1])) * 16 + quant16(p[2]);
            atomicAdd(&lhist[bb], 1u);
        }
    }
    __syncthreads();
    for (int i = threadIdx.x; i < NBINS; i += blockDim.x) {
        unsigned int v = lhist[i];
        if (v) atomicAdd(&ghist[i], v);                 // global_atomic_add_u32
    }
}

// Target palette histogram: tiny (4096 points) — single block.
__global__ __launch_bounds__(256)
void hist_tgt_kernel(const float* __restrict__ colors,
                     unsigned int* __restrict__ ghist, int npoints) {
    __shared__ unsigned int lhist[NBINS];
    for (int i = threadIdx.x; i < NBINS; i += 256) lhist[i] = 0u;
    __syncthreads();
    for (int i = threadIdx.x; i < npoints; i += 256) {
        const float* p = colors + (size_t)i * 3;
        int bb = (quant16(p[0]) * 16 + quant16(p[1])) * 16 + quant16(p[2]);
        atomicAdd(&lhist[bb], 1u);
    }
    __syncthreads();
    for (int i = threadIdx.x; i < NBINS; i += 256) ghist[i] = lhist[i];  // sole writer
}

// Final loss: per-lane partial sums of |s/Ns - t/Nt|, wave-level reduction via
// v_wmma_f32_16x16x4_f32 with an all-ones A matrix (D's every row == column
// sums of B, so the result is robust to exact B lane striping), then 4
// shfl_xor steps across the 16 columns, then one LDS atomic across 8 waves.
__global__ __launch_bounds__(256)
void loss_kernel(const unsigned int* __restrict__ hs,
                 const unsigned int* __restrict__ ht,
                 float invNs, float invNt, float* __restrict__ out) {
    __shared__ float acc;
    if (threadIdx.x == 0) acc = 0.0f;
    __syncthreads();

    float partial = 0.0f;
#pragma unroll
    for (int k = 0; k < NBINS / 256; ++k) {
        int i = threadIdx.x + k * 256;
        float s = (float)hs[i] * invNs;
        float t = (float)ht[i] * invNt;
        partial += fabsf(s - t);
    }

    // WMMA reduction: A = ones(16x4), B holds each lane's partial (b.y = 0).
    // sum over all B elements == sum of partials; EXEC is all-ones here.
    v2f a; a.x = 1.0f; a.y = 1.0f;
    v2f b; b.x = partial; b.y = 0.0f;
    v8f c = {};
    v8f d = __builtin_amdgcn_wmma_f32_16x16x4_f32(
        /*neg_a=*/false, a, /*neg_b=*/false, b,
        /*c_mod=*/(short)0, c, /*reuse_a=*/false, /*reuse_b=*/false);

    float r = d[0];                        // every lane: column sum of its N=lane%16
    r += __shfl_xor(r, 1, 32);
    r += __shfl_xor(r, 2, 32);
    r += __shfl_xor(r, 4, 32);
    r += __shfl_xor(r, 8, 32);             // r == wave total (all lanes)

    if ((threadIdx.x & 31) == 0) atomicAdd(&acc, r);
    __syncthreads();
    if (threadIdx.x == 0) out[0] = acc * (1.0f / (float)NBINS);
}

extern "C" void kernel_launch(void* const* d_in, const int* in_sizes, int n_in,
                              void* d_out, int out_size, void* d_ws, size_t ws_size,
                              hipStream_t stream) {
    const float* src = (const float*)d_in[0];
    const float* tgt = (const float*)d_in[1];
    const int n_src = in_sizes[0] / 3;     // 8388608
    const int n_tgt = in_sizes[1] / 3;     // 4096

    unsigned int* hs = (unsigned int*)d_ws;
    unsigned int* ht = hs + NBINS;

    zero_hist_kernel<<<(2 * NBINS + 255) / 256, 256, 0, stream>>>(hs, 2 * NBINS);
    hist_src_kernel<<<192, 1024, 0, stream>>>(src, hs, n_src);
    hist_tgt_kernel<<<1, 256, 0, stream>>>(tgt, ht, n_tgt);
    loss_kernel<<<1, 256, 0, stream>>>(hs, ht,
                                       1.0f / (float)n_src, 1.0f / (float)n_tgt,
                                       (float*)d_out);
}